// MeshRasterizer_72507637891072
// MI455X (gfx1250) — compile-verified
//
#include <hip/hip_runtime.h>
#include <math.h>

constexpr int   kH = 128;
constexpr int   kB = 2;
constexpr int   kV = 2048;
constexpr int   kF = 512;
constexpr float kFocal = 3.0f;
constexpr float kZNear = 0.01f;
constexpr float kEps   = 1e-8f;

typedef float v2f __attribute__((ext_vector_type(2)));
typedef float v8f __attribute__((ext_vector_type(8)));
typedef int   v2i_g __attribute__((vector_size(2 * sizeof(int))));

#define AS1 __attribute__((address_space(1)))
#define AS3 __attribute__((address_space(3)))

#if defined(__has_builtin)
#if __has_builtin(__builtin_amdgcn_global_load_async_to_lds_b64) && \
    __has_builtin(__builtin_amdgcn_global_load_async_to_lds_b32)
#define HAVE_ASYNC_LDS 1
#endif
#endif

__device__ __forceinline__ void wait_async0() {
#if defined(__has_builtin) && __has_builtin(__builtin_amdgcn_s_wait_asynccnt)
    __builtin_amdgcn_s_wait_asynccnt(0);
#else
    asm volatile("s_wait_asynccnt 0" ::: "memory");
#endif
}

// ---------------------------------------------------------------------------
// Kernel 1: project vertices to NDC:  v_view = verts @ R + T ; xy = f*xy/z
// ---------------------------------------------------------------------------
__global__ void project_kernel(const float* __restrict__ verts,
                               const float* __restrict__ Rm,
                               const float* __restrict__ Tm,
                               float* __restrict__ vndc) {
    int i = blockIdx.x * blockDim.x + threadIdx.x;
    if (i >= kB * kV) return;
    int b = i / kV;
    const float* p = verts + i * 3;
    const float* R = Rm + b * 9;
    const float* T = Tm + b * 3;
    float x = p[0], y = p[1], z = p[2];
    float vx = fmaf(x, R[0], fmaf(y, R[3], fmaf(z, R[6], T[0])));
    float vy = fmaf(x, R[1], fmaf(y, R[4], fmaf(z, R[7], T[1])));
    float vz = fmaf(x, R[2], fmaf(y, R[5], fmaf(z, R[8], T[2])));
    float* o = vndc + i * 3;
    o[0] = kFocal * vx / vz;
    o[1] = kFocal * vy / vz;
    o[2] = vz;
}

// ---------------------------------------------------------------------------
// Kernel 2: per-face affine coefficients.
//  fcPair[b][q][f] (float2), q: 0=(a0,b0) 1=(c0,0) 2=(a1,b1) 3=(c1,0)
//                               4=(az,bz) 5=(cz,0)
//  fcValid[b][f] : |area|>eps as 0/1 ;  fcVerts[b][q][f], q=x0,y0,x1,y1,x2,y2
// ---------------------------------------------------------------------------
__global__ void face_setup_kernel(const int* __restrict__ faces,
                                  const float* __restrict__ vndc,
                                  float2* __restrict__ fcPair,
                                  float* __restrict__ fcValid,
                                  float* __restrict__ fcVerts) {
    int i = blockIdx.x * blockDim.x + threadIdx.x;
    if (i >= kB * kF) return;
    int b = i / kF, f = i % kF;
    int i0 = faces[f * 3 + 0], i1 = faces[f * 3 + 1], i2 = faces[f * 3 + 2];
    const float* vb = vndc + b * kV * 3;
    float x0 = vb[i0 * 3 + 0], y0 = vb[i0 * 3 + 1], z0 = vb[i0 * 3 + 2];
    float x1 = vb[i1 * 3 + 0], y1 = vb[i1 * 3 + 1], z1 = vb[i1 * 3 + 2];
    float x2 = vb[i2 * 3 + 0], y2 = vb[i2 * 3 + 1], z2 = vb[i2 * 3 + 2];

    float area  = (x1 - x0) * (y2 - y0) - (y1 - y0) * (x2 - x0);
    float denom = (fabsf(area) > kEps) ? area : kEps;
    float inv   = 1.0f / denom;

    float a0 = -(y2 - y1) * inv;
    float b0 =  (x2 - x1) * inv;
    float c0 = ((y2 - y1) * x1 - (x2 - x1) * y1) * inv;
    float a1 = -(y0 - y2) * inv;
    float b1 =  (x0 - x2) * inv;
    float c1 = ((y0 - y2) * x2 - (x0 - x2) * y2) * inv;
    float dz0 = z0 - z2, dz1 = z1 - z2;
    float az = a0 * dz0 + a1 * dz1;
    float bz = b0 * dz0 + b1 * dz1;
    float cz = c0 * dz0 + c1 * dz1 + z2;

    int pb = b * 6 * kF;
    fcPair[pb + 0 * kF + f] = make_float2(a0, b0);
    fcPair[pb + 1 * kF + f] = make_float2(c0, 0.0f);
    fcPair[pb + 2 * kF + f] = make_float2(a1, b1);
    fcPair[pb + 3 * kF + f] = make_float2(c1, 0.0f);
    fcPair[pb + 4 * kF + f] = make_float2(az, bz);
    fcPair[pb + 5 * kF + f] = make_float2(cz, 0.0f);
    fcValid[b * kF + f] = (fabsf(area) > kEps) ? 1.0f : 0.0f;
    int vb6 = b * 6 * kF;
    fcVerts[vb6 + 0 * kF + f] = x0;  fcVerts[vb6 + 1 * kF + f] = y0;
    fcVerts[vb6 + 2 * kF + f] = x1;  fcVerts[vb6 + 3 * kF + f] = y1;
    fcVerts[vb6 + 4 * kF + f] = x2;  fcVerts[vb6 + 5 * kF + f] = y2;
}

// ---------------------------------------------------------------------------
// Kernel 3: WMMA rasterizer. One wave = 16 pixels of a row; 32 face tiles.
// Coefficients staged block-wide into LDS via async-to-LDS (ASYNCcnt).
// ---------------------------------------------------------------------------
__device__ __forceinline__ float seg_d2(float px, float py, float ax, float ay,
                                        float bx, float by) {
    float abx = bx - ax, aby = by - ay;
    float apx = px - ax, apy = py - ay;
    float t = (apx * abx + apy * aby) / (abx * abx + aby * aby + kEps);
    t = fminf(fmaxf(t, 0.0f), 1.0f);
    float dx = apx - t * abx, dy = apy - t * aby;
    return dx * dx + dy * dy;
}

__global__ void __launch_bounds__(256)
rasterize_kernel(const float2* __restrict__ fcPair,
                 const float* __restrict__ fcValid,
                 const float* __restrict__ fcVerts,
                 float* __restrict__ out) {
    constexpr int HH = kH * kH;
    __shared__ float2 sPair[6 * kF];   // 24 KB
    __shared__ float  sValid[kF];      //  2 KB

    int wave = threadIdx.x >> 5;
    int lane = threadIdx.x & 31;
    int tile = blockIdx.x * 8 + wave;      // 0..2047
    int b    = blockIdx.x >> 7;            // uniform per block
    int rem  = tile & 1023;
    int y    = rem >> 3;                    // image row
    int tx   = rem & 7;                     // 16-pixel tile in row
    int n    = lane & 15;
    bool hi  = lane >= 16;

    // ---- stage coefficient table for this batch into LDS (async path) ----
    const float2* gPair  = fcPair + b * 6 * kF;
    const float*  gValid = fcValid + b * kF;
    const float*  gVerts = fcVerts + b * 6 * kF;
    __builtin_prefetch(gVerts, 0, 0);       // winner-only gathers later
#ifdef HAVE_ASYNC_LDS
    for (int i = threadIdx.x; i < 6 * kF; i += 256)
        __builtin_amdgcn_global_load_async_to_lds_b64(
            (AS1 v2i_g*)(gPair + i), (AS3 v2i_g*)(sPair + i), 0, 0);
    for (int i = threadIdx.x; i < kF; i += 256)
        __builtin_amdgcn_global_load_async_to_lds_b32(
            (AS1 int*)(gValid + i), (AS3 int*)(sValid + i), 0, 0);
    wait_async0();
#else
    for (int i = threadIdx.x; i < 6 * kF; i += 256) sPair[i] = gPair[i];
    for (int i = threadIdx.x; i < kF; i += 256)     sValid[i] = gValid[i];
#endif
    __syncthreads();

    // ---- A operand (16x4 f32): lanes 0-15 K=0,1 ; lanes 16-31 K=2,3 ----
    float s_py = 1.0f - (2.0f * y + 1.0f) / kH;
    float s_px = 1.0f - (2.0f * (tx * 16 + n) + 1.0f) / kH;
    v2f amat;
    amat.x = hi ? 1.0f : s_px;   // K0 = px , K2 = 1
    amat.y = hi ? 0.0f : s_py;   // K1 = py , K3 = 0

    // ---- packed argmin key per C-slot: (z_bits<<32)|face, min == argmin ----
    unsigned long long bestKey[8];
#pragma unroll
    for (int r = 0; r < 8; ++r) bestKey[r] = ~0ull;

    int qoff = hi ? 1 : 0;    // lanes 0-15 take (a,b) rows, 16-31 take (c,0)
    for (int ft = 0; ft < kF / 16; ++ft) {
        int f = ft * 16 + n;
        float2 p0 = sPair[(0 + qoff) * kF + f];
        float2 p1 = sPair[(2 + qoff) * kF + f];
        float2 pz = sPair[(4 + qoff) * kF + f];
        float fvalid = sValid[f];

        v2f b0v; b0v.x = p0.x; b0v.y = p0.y;
        v2f b1v; b1v.x = p1.x; b1v.y = p1.y;
        v2f bzv; bzv.x = pz.x; bzv.y = pz.y;

        v8f czero = {};
        v8f w0m = __builtin_amdgcn_wmma_f32_16x16x4_f32(
            false, amat, false, b0v, (short)0, czero, false, false);
        v8f w1m = __builtin_amdgcn_wmma_f32_16x16x4_f32(
            false, amat, false, b1v, (short)0, czero, false, false);
        v8f zm  = __builtin_amdgcn_wmma_f32_16x16x4_f32(
            false, amat, false, bzv, (short)0, czero, false, false);

#pragma unroll
        for (int r = 0; r < 8; ++r) {
            float w0 = w0m[r], w1 = w1m[r];
            float w2 = 1.0f - w0 - w1;
            float zp = zm[r];
            bool valid = (w0 >= 0.0f) && (w1 >= 0.0f) && (w2 >= 0.0f) &&
                         (zp > kZNear) && (fvalid > 0.5f);
            float zs = valid ? zp : __builtin_inff();
            unsigned long long key =
                ((unsigned long long)__float_as_uint(zs) << 32) | (unsigned)f;
            bestKey[r] = key < bestKey[r] ? key : bestKey[r];   // branchless
        }
    }

    // ---- reduce over the 16 face-columns (within each 16-lane half) ----
#pragma unroll
    for (int r = 0; r < 8; ++r) {
        unsigned long long bk = bestKey[r];
        for (int m = 1; m < 16; m <<= 1) {
            unsigned long long ok = __shfl_xor(bk, m, 16);
            bk = ok < bk ? ok : bk;
        }
        bestKey[r] = bk;
    }

    // ---- outputs: [p2f | zbuf | bary*3 | dist] ----
    float* p2f = out;
    float* zb  = out + kB * HH;
    float* bar = out + 2 * kB * HH;
    float* ds  = out + 5 * kB * HH;

#pragma unroll
    for (int r = 0; r < 8; ++r) {
        int m  = r + (hi ? 8 : 0);
        int gp = b * HH + y * kH + tx * 16 + m;
        float pxm = 1.0f - (2.0f * (tx * 16 + m) + 1.0f) / kH;

        unsigned zbits = (unsigned)(bestKey[r] >> 32);
        bool hit = zbits < 0x7f800000u;            // uniform within half-wave
        float o_p2f = -1.0f, o_z = -1.0f, o_d = -1.0f;
        float o_b0 = -1.0f, o_b1 = -1.0f, o_b2 = -1.0f;
        if (hit) {
            int f = (int)(unsigned)(bestKey[r] & 0xffffffffu);
            // winner's barycentrics from its affine coefficients
            float2 q0 = sPair[0 * kF + f]; float c0 = sPair[1 * kF + f].x;
            float2 q1 = sPair[2 * kF + f]; float c1 = sPair[3 * kF + f].x;
            float w0 = fmaf(q0.x, pxm, fmaf(q0.y, s_py, c0));
            float w1 = fmaf(q1.x, pxm, fmaf(q1.y, s_py, c1));
            float x0 = gVerts[0 * kF + f], y0 = gVerts[1 * kF + f];
            float x1 = gVerts[2 * kF + f], y1 = gVerts[3 * kF + f];
            float x2 = gVerts[4 * kF + f], y2 = gVerts[5 * kF + f];
            float d2 = fminf(seg_d2(pxm, s_py, x0, y0, x1, y1),
                       fminf(seg_d2(pxm, s_py, x1, y1, x2, y2),
                             seg_d2(pxm, s_py, x2, y2, x0, y0)));
            o_p2f = (float)f;
            o_z   = __uint_as_float(zbits);        // exact WMMA zp of winner
            o_b0  = w0;
            o_b1  = w1;
            o_b2  = 1.0f - w0 - w1;
            o_d   = -d2;                           // winner is inside => -d2
        }
        if (n == 0) {                              // one writer per half-wave
            p2f[gp] = o_p2f;
            zb[gp]  = o_z;
            ds[gp]  = o_d;
            bar[gp * 3 + 0] = o_b0;
            bar[gp * 3 + 1] = o_b1;
            bar[gp * 3 + 2] = o_b2;
        }
    }
}

// ---------------------------------------------------------------------------
extern "C" void kernel_launch(void* const* d_in, const int* in_sizes, int n_in,
                              void* d_out, int out_size, void* d_ws, size_t ws_size,
                              hipStream_t stream) {
    const float* verts = (const float*)d_in[0];   // [B,V,3] f32
    const int*   faces = (const int*)d_in[1];     // [F,3]   i32
    const float* Rm    = (const float*)d_in[2];   // [B,3,3] f32
    const float* Tm    = (const float*)d_in[3];   // [B,3]   f32
    float* out = (float*)d_out;

    float* ws      = (float*)d_ws;
    float* vndc    = ws;                              // kB*kV*3      = 12288 f
    float* pairF   = vndc + kB * kV * 3;              // kB*6*kF*2    = 12288 f
    float* validF  = pairF + kB * 6 * kF * 2;         // kB*kF        =  1024 f
    float* vertsF  = validF + kB * kF;                // kB*6*kF      =  6144 f

    project_kernel<<<(kB * kV + 255) / 256, 256, 0, stream>>>(verts, Rm, Tm, vndc);
    face_setup_kernel<<<(kB * kF + 255) / 256, 256, 0, stream>>>(
        faces, vndc, (float2*)pairF, validF, vertsF);
    rasterize_kernel<<<256, 256, 0, stream>>>(
        (const float2*)pairF, validF, vertsF, out);
}